// PMPModel_46377056862941
// MI455X (gfx1250) — compile-verified
//
#include <hip/hip_runtime.h>
#include <hip/hip_bf16.h>
#include <math.h>

typedef __attribute__((ext_vector_type(2))) float v2f;
typedef __attribute__((ext_vector_type(8))) float v8f;

#define DIM 128
#define LDS_STRIDE 132   // 128 + 4 pad: conflict-free column (A-fragment) reads
#define NODES_PER_BLOCK 64

// ---------------------------------------------------------------------------
// One 16x16 output tile: acc = Xtile(16x128) @ W(:,col0:col0+16), K=128 via
// 32 x V_WMMA_F32_16X16X4_F32. Stores are UNGUARDED (workspace rows padded
// to a multiple of 64); bias emitted only when BIAS (no +0.0 adds otherwise).
// ---------------------------------------------------------------------------
template <bool BIAS>
__device__ __forceinline__ void tile_gemm_f32(
    const float* __restrict__ lds_row,   // &lds_x[(rbl + (lane&15)) * LDS_STRIDE]
    const float* __restrict__ W,         // [128][128] row-major (k, out)
    float*       __restrict__ O,         // padded [Npad][128]
    const float* __restrict__ bias,      // [128] (used iff BIAS)
    int n0, int rbl, int half, int l15, int col0)
{
    const int kOff = half << 1;          // lanes 0-15 -> K 0,1 ; lanes 16-31 -> K 2,3
    v8f acc = {0.f, 0.f, 0.f, 0.f, 0.f, 0.f, 0.f, 0.f};

    #pragma unroll 8
    for (int k0 = 0; k0 < DIM; k0 += 4) {
        const int kA = k0 + kOff;
        v2f a; a.x = lds_row[kA]; a.y = lds_row[kA + 1];
        const float* wr = W + (size_t)kA * DIM + col0 + l15;
        v2f b; b.x = wr[0]; b.y = wr[DIM];
        acc = __builtin_amdgcn_wmma_f32_16x16x4_f32(
                false, a, false, b, (short)0, acc, false, false);
    }

    // D layout: VGPR r -> row M=r (lanes 0-15) / M=r+8 (lanes 16-31)
    float* op = O + (size_t)(n0 + rbl + (half << 3)) * DIM + col0 + l15;
    if (BIAS) {
        const float bv = bias[col0 + l15];
        #pragma unroll
        for (int r = 0; r < 8; ++r)
            op[(size_t)r * DIM] = acc[r] + bv;
    } else {
        #pragma unroll
        for (int r = 0; r < 8; ++r)
            op[(size_t)r * DIM] = acc[r];
    }
}

// ---------------------------------------------------------------------------
// Node kernel (64-node tile, 256 threads = 8 waves):
//   fr   = X @ W_fr;  be = X @ W_be;  aggr = X @ Wself + bself
//   alpha= sigmoid(X @ aw + ab)
// Wave w: rows (w>>1)*16..+15, cols (w&1)*64..+63 (4 col-tiles).
// ---------------------------------------------------------------------------
template <bool RELU_IN>
__global__ __launch_bounds__(256)
void pmp_node_kernel(const float* __restrict__ X,
                     const float* __restrict__ W_fr,
                     const float* __restrict__ W_be,
                     const float* __restrict__ W_self,
                     const float* __restrict__ b_self,
                     const float* __restrict__ alpha_w,
                     const float* __restrict__ alpha_b,
                     float* __restrict__ fr,
                     float* __restrict__ be,
                     float* __restrict__ aggr,
                     float* __restrict__ alpha,
                     int N)
{
    __shared__ float lds_x[NODES_PER_BLOCK * LDS_STRIDE];

    const int tid = threadIdx.x;
    const int n0  = blockIdx.x * NODES_PER_BLOCK;

    // ---- stage X tile into LDS (float4, zero-pad tail rows, optional ReLU) ----
    {
        const float4 zero4 = make_float4(0.f, 0.f, 0.f, 0.f);
        #pragma unroll
        for (int i = 0; i < 8; ++i) {                 // 2048 float4 / 256 threads
            int fe   = i * 256 + tid;
            int row  = fe >> 5;
            int c4   = fe & 31;
            int node = n0 + row;
            float4 v = (node < N) ? ((const float4*)(X + (size_t)node * DIM))[c4] : zero4;
            if (RELU_IN) {
                v.x = fmaxf(v.x, 0.f); v.y = fmaxf(v.y, 0.f);
                v.z = fmaxf(v.z, 0.f); v.w = fmaxf(v.w, 0.f);
            }
            *(float4*)&lds_x[row * LDS_STRIDE + c4 * 4] = v;
        }
    }
    __syncthreads();

    // ---- alpha GEMV + sigmoid (threads 0..63, one node each) ----
    if (tid < NODES_PER_BLOCK) {
        int node = n0 + tid;
        if (node < N) {
            float acc = alpha_b[0];
            const float* xr = &lds_x[tid * LDS_STRIDE];
            #pragma unroll 8
            for (int k = 0; k < DIM; ++k) acc += xr[k] * alpha_w[k];
            alpha[node] = 1.f / (1.f + __expf(-acc));
        }
    }

    // ---- WMMA FP32 16x16x4 main loop ----
    const int wave = tid >> 5;
    const int lane = tid & 31;
    const int half = lane >> 4;
    const int l15  = lane & 15;
    const int rbl  = (wave >> 1) * 16;     // 0,16,32,48
    const int ctb  = (wave & 1) * 4;       // col-tile base

    const float* lds_row = &lds_x[(rbl + l15) * LDS_STRIDE];

    #pragma unroll
    for (int ct = 0; ct < 4; ++ct) {
        const int col0 = (ctb + ct) * 16;
        tile_gemm_f32<false>(lds_row, W_fr,   fr,   nullptr, n0, rbl, half, l15, col0);
        tile_gemm_f32<false>(lds_row, W_be,   be,   nullptr, n0, rbl, half, l15, col0);
        tile_gemm_f32<true >(lds_row, W_self, aggr, b_self,  n0, rbl, half, l15, col0);
    }
}

// ---------------------------------------------------------------------------
// Edge kernel: one wave32 per edge; lane handles 4 consecutive features.
//   nt==1 -> fr[src];  nt==0 -> be[src];  nt==2 -> be + a*(fr-be), a=alpha[dst]
// ---------------------------------------------------------------------------
__global__ __launch_bounds__(256)
void pmp_edge_kernel(const int* __restrict__ src_arr,
                     const int* __restrict__ dst_arr,
                     const int* __restrict__ y,
                     const int* __restrict__ pmp_mask,
                     const float* __restrict__ fr,
                     const float* __restrict__ be,
                     const float* __restrict__ alpha,
                     float* __restrict__ aggr,
                     int E)
{
    const int gid  = blockIdx.x * blockDim.x + threadIdx.x;
    const int edge = gid >> 5;
    const int lane = gid & 31;
    if (edge >= E) return;

    const int s  = src_arr[edge];
    const int d  = dst_arr[edge];
    const int nt = pmp_mask[s] ? y[s] : 2;

    const float4* frp = (const float4*)(fr + (size_t)s * DIM);
    const float4* bep = (const float4*)(be + (size_t)s * DIM);

    float4 v;
    if (nt == 1) {                         // wave-uniform branches (nt is per-edge)
        v = frp[lane];
    } else if (nt == 0) {
        v = bep[lane];
    } else {
        const float a = alpha[d];
        float4 f = frp[lane];
        float4 b = bep[lane];
        v.x = b.x + a * (f.x - b.x);
        v.y = b.y + a * (f.y - b.y);
        v.z = b.z + a * (f.z - b.z);
        v.w = b.w + a * (f.w - b.w);
    }

    float* out = aggr + (size_t)d * DIM + lane * 4;
    atomicAdd(out + 0, v.x);
    atomicAdd(out + 1, v.y);
    atomicAdd(out + 2, v.z);
    atomicAdd(out + 3, v.w);
}

// ---------------------------------------------------------------------------
// Classifier: out[n][o] = relu(h[n]) . cls_w[:,o] + cls_b[o]   (OUT = 8)
// ---------------------------------------------------------------------------
__global__ __launch_bounds__(256)
void pmp_cls_kernel(const float* __restrict__ h,
                    const float* __restrict__ cls_w,
                    const float* __restrict__ cls_b,
                    float* __restrict__ out,
                    int N)
{
    const int gid = blockIdx.x * blockDim.x + threadIdx.x;
    if (gid >= N * 8) return;
    const int n = gid >> 3;
    const int o = gid & 7;
    const float* hr = h + (size_t)n * DIM;
    float acc = cls_b[o];
    #pragma unroll 8
    for (int k = 0; k < DIM; ++k)
        acc += fmaxf(hr[k], 0.f) * cls_w[k * 8 + o];
    out[gid] = acc;
}

// ---------------------------------------------------------------------------
extern "C" void kernel_launch(void* const* d_in, const int* in_sizes, int n_in,
                              void* d_out, int out_size, void* d_ws, size_t ws_size,
                              hipStream_t stream)
{
    // setup_inputs() order
    const float* x        = (const float*)d_in[0];
    const int*   ei       = (const int*)  d_in[1];   // [2,E] -> src=ei, dst=ei+E
    const int*   y        = (const int*)  d_in[2];
    const int*   pmp_mask = (const int*)  d_in[3];
    const float* W_fr1    = (const float*)d_in[4];
    const float* W_be1    = (const float*)d_in[5];
    const float* aw1      = (const float*)d_in[6];
    const float* ab1      = (const float*)d_in[7];
    const float* Wself1   = (const float*)d_in[8];
    const float* bself1   = (const float*)d_in[9];
    const float* W_fr2    = (const float*)d_in[10];
    const float* W_be2    = (const float*)d_in[11];
    const float* aw2      = (const float*)d_in[12];
    const float* ab2      = (const float*)d_in[13];
    const float* Wself2   = (const float*)d_in[14];
    const float* bself2   = (const float*)d_in[15];
    const float* cls_w    = (const float*)d_in[16];
    const float* cls_b    = (const float*)d_in[17];

    const int N = in_sizes[0] / DIM;
    const int E = in_sizes[1] / 2;
    const int* src = ei;
    const int* dst = ei + E;

    const int node_blocks = (N + NODES_PER_BLOCK - 1) / NODES_PER_BLOCK;
    const int Npad        = node_blocks * NODES_PER_BLOCK;   // padded rows: unguarded stores

    // workspace layout (fp32), 256B aligned padded slabs
    const size_t slab = ((size_t)Npad * DIM * sizeof(float) + 255) & ~(size_t)255;
    char* ws = (char*)d_ws;
    float* FR    = (float*)(ws);
    float* BE    = (float*)(ws + slab);
    float* AGG1  = (float*)(ws + 2 * slab);
    float* AGG2  = (float*)(ws + 3 * slab);
    float* ALPHA = (float*)(ws + 4 * slab);

    const int edge_blocks = (int)(((size_t)E * 32 + 255) / 256);
    const int cls_blocks  = (N * 8 + 255) / 256;

    // ---- layer 1 ----
    pmp_node_kernel<false><<<node_blocks, 256, 0, stream>>>(
        x, W_fr1, W_be1, Wself1, bself1, aw1, ab1, FR, BE, AGG1, ALPHA, N);
    pmp_edge_kernel<<<edge_blocks, 256, 0, stream>>>(
        src, dst, y, pmp_mask, FR, BE, ALPHA, AGG1, E);

    // ---- layer 2 (ReLU applied on load of AGG1) ----
    pmp_node_kernel<true><<<node_blocks, 256, 0, stream>>>(
        AGG1, W_fr2, W_be2, Wself2, bself2, aw2, ab2, FR, BE, AGG2, ALPHA, N);
    pmp_edge_kernel<<<edge_blocks, 256, 0, stream>>>(
        src, dst, y, pmp_mask, FR, BE, ALPHA, AGG2, E);

    // ---- classifier (ReLU on load of AGG2) ----
    pmp_cls_kernel<<<cls_blocks, 256, 0, stream>>>(
        AGG2, cls_w, cls_b, (float*)d_out, N);
}